// Detect_26714696581697
// MI455X (gfx1250) — compile-verified
//
#include <hip/hip_runtime.h>
#include <stdint.h>

#define NUM_CLASSES 21
#define TOPK        100
#define P_PRIORS    25500
#define BATCH       16
#define CONF_THR    0.3f
#define NMS_THR     0.45f
#define POS_PRIOR_THR 0.01f
#define NEGV        (-1e9f)

#ifndef __has_builtin
#define __has_builtin(x) 0
#endif

#if __has_builtin(__builtin_amdgcn_tensor_load_to_lds)
#define USE_TDM 1
#else
#define USE_TDM 0
#endif

typedef unsigned int u32x4 __attribute__((ext_vector_type(4)));
typedef int          i32x8 __attribute__((ext_vector_type(8)));
typedef int          i32x4 __attribute__((ext_vector_type(4)));

// ---------- helpers ----------

// order-preserving float->uint mapping (monotone increasing)
__device__ __forceinline__ unsigned flip_f32(float f) {
  unsigned u = __float_as_uint(f);
  return (u & 0x80000000u) ? ~u : (u ^ 0x80000000u);
}
__device__ __forceinline__ float unflip_f32(unsigned u) {
  unsigned b = (u & 0x80000000u) ? (u ^ 0x80000000u) : ~u;
  return __uint_as_float(b);
}

// write v into bit range [lo, lo+width) of a dword array (uniform/scalar use)
__device__ __forceinline__ void set_bits(unsigned* w, int lo, int width,
                                         unsigned long long v) {
  int i = 0;
  while (i < width) {
    int d = (lo + i) >> 5;
    int b = (lo + i) & 31;
    int n = 32 - b; if (n > width - i) n = width - i;
    unsigned long long fm = (n == 32) ? 0xFFFFFFFFull : ((1ull << n) - 1ull);
    unsigned mask = (unsigned)(fm << b);
    unsigned val  = (unsigned)(((v >> i) & fm) << b);
    w[d] = (w[d] & ~mask) | val;
    i += n;
  }
}

#if USE_TDM
// Build a D# for a 2D tile (tile0 x tile1 elements) and issue TENSOR_LOAD_TO_LDS.
// data_size_code: 0=1B, 1=2B, 2=4B, 3=8B.  stride0 in elements.
__device__ __forceinline__ void tdm_load_2d(unsigned lds_off, const void* gaddr,
                                            unsigned data_size_code,
                                            unsigned tile0, unsigned tile1,
                                            unsigned long long tdim0,
                                            unsigned long long tdim1,
                                            unsigned long long stride0) {
  unsigned g0[4] = {0, 0, 0, 0};
  unsigned g1[8] = {0, 0, 0, 0, 0, 0, 0, 0};
  unsigned long long ga = (unsigned long long)(uintptr_t)gaddr;
  g0[0] = 1u;                                   // count=1, gather off, user mode
  g0[1] = lds_off;                              // lds_addr (bytes)
  g0[2] = (unsigned)(ga & 0xFFFFFFFFull);       // global_addr[31:0]
  g0[3] = (unsigned)((ga >> 32) & 0x1FFFFFFull) // global_addr[56:32]
        | (2u << 30);                           // type = 2 ("image")
  set_bits(g1, 16, 2,  data_size_code);         // data_size
  set_bits(g1, 48, 32, tdim0);                  // tensor_dim0
  set_bits(g1, 80, 32, tdim1);                  // tensor_dim1
  set_bits(g1, 112, 16, tile0);                 // tile_dim0
  set_bits(g1, 128, 16, tile1);                 // tile_dim1
  set_bits(g1, 160, 48, stride0);               // tensor_dim0_stride
  u32x4 v0 = {g0[0], g0[1], g0[2], g0[3]};
  i32x8 v1 = {(int)g1[0], (int)g1[1], (int)g1[2], (int)g1[3],
              (int)g1[4], (int)g1[5], (int)g1[6], (int)g1[7]};
  i32x4 z4 = {0, 0, 0, 0};
  i32x8 z8 = {0, 0, 0, 0, 0, 0, 0, 0};
  // clang-23 / therock-10.0 toolchain: 6-arg form
  __builtin_amdgcn_tensor_load_to_lds(v0, v1, z4, z4, z8, 0);
}
#endif

// ---------- kernel A: ARM positive-prior mask ----------
__global__ void arm_mask_kernel(const float* __restrict__ bi_conf,
                                unsigned char* __restrict__ mask, int n) {
  int i = blockIdx.x * blockDim.x + threadIdx.x;
  if (i >= n) return;
  float c0 = bi_conf[2 * (size_t)i];
  float c1 = bi_conf[2 * (size_t)i + 1];
  float armpos = 1.0f / (1.0f + expf(c0 - c1));   // softmax()[...,1]
  mask[i] = (armpos >= POS_PRIOR_THR) ? 1 : 0;
}

// ---------- kernel B: per-(b,c) top-k + decode + NMS ----------
__global__ __launch_bounds__(256)
void detect_topk_nms_kernel(const float* __restrict__ bi_loc,
                            const float* __restrict__ multi_loc,
                            const float* __restrict__ multi_conf,
                            const float* __restrict__ prior,
                            const unsigned char* __restrict__ arm_mask,
                            float* __restrict__ out) {
  const int c = blockIdx.x;
  const int b = blockIdx.y;
  const int t = threadIdx.x;
  float* plane = out + (((size_t)b * NUM_CLASSES + c) * TOPK) * 5;

  if (c == 0) {  // background class: whole plane zero
    for (int i = t; i < TOPK * 5; i += 256) plane[i] = 0.0f;
    return;
  }

  __shared__ float              s_conf[P_PRIORS + 4];
  __shared__ unsigned char      s_mask[P_PRIORS + 4];
  __shared__ unsigned long long s_part[256];
  __shared__ float              s_selScore[TOPK];
  __shared__ int                s_selIdx[TOPK];
  __shared__ float              s_box[TOPK][4];
  __shared__ float              s_area[TOPK];
  __shared__ int                s_alive[TOPK];

  const float* colbase = multi_conf + (size_t)b * P_PRIORS * NUM_CLASSES + c;
  const unsigned char* mrow = arm_mask + (size_t)b * P_PRIORS;

#if USE_TDM
  if (t == 0) {
    // strided column gather: tile 1 x P, stride 21 elements of 4B -> contiguous LDS
    tdm_load_2d((unsigned)(uintptr_t)(void*)s_conf, colbase,
                /*4B*/2, /*tile0*/1, /*tile1*/P_PRIORS,
                /*tdim0*/NUM_CLASSES, /*tdim1*/P_PRIORS, /*stride0*/NUM_CLASSES);
    // contiguous byte row: tile P x 1
    tdm_load_2d((unsigned)(uintptr_t)(void*)s_mask, mrow,
                /*1B*/0, /*tile0*/P_PRIORS, /*tile1*/1,
                /*tdim0*/P_PRIORS, /*tdim1*/1, /*stride0*/P_PRIORS);
#if __has_builtin(__builtin_amdgcn_s_wait_tensorcnt)
    __builtin_amdgcn_s_wait_tensorcnt(0);
#else
    asm volatile("s_wait_tensorcnt 0x0" ::: "memory");
#endif
  }
#else
  for (int i = t; i < P_PRIORS; i += 256) s_conf[i] = colbase[(size_t)i * NUM_CLASSES];
  for (int i = t; i < P_PRIORS; i += 256) s_mask[i] = mrow[i];
#endif
  __syncthreads();

  // ---- per-thread partial max over a contiguous 100-element chunk ----
  const int CH = 100;
  const int p0 = t * CH;
  const int p1 = (p0 + CH < P_PRIORS) ? (p0 + CH) : P_PRIORS;
  unsigned long long myMax = 0ull;
  for (int p = p0; p < p1; ++p) {
    float v = s_conf[p];
    float s = (s_mask[p] && v > CONF_THR) ? v : NEGV;
    unsigned long long k = ((unsigned long long)flip_f32(s) << 15) |
                           (unsigned long long)(P_PRIORS - p);   // tie -> lower p
    if (k > myMax) myMax = k;
  }

  // ---- exact top-100 via iterated block argmax ----
  for (int k = 0; k < TOPK; ++k) {
    s_part[t] = myMax;
    __syncthreads();
    for (int off = 128; off > 0; off >>= 1) {
      if (t < off) { if (s_part[t + off] > s_part[t]) s_part[t] = s_part[t + off]; }
      __syncthreads();
    }
    unsigned long long w = s_part[0];
    int wp = P_PRIORS - (int)(w & 0x7FFFull);
    if (t == 0) {
      s_selScore[k] = unflip_f32((unsigned)(w >> 15));
      s_selIdx[k] = wp;
    }
    __syncthreads();                    // everyone has read s_part[0]
    if (wp >= p0 && wp < p1) {          // owning thread removes winner, rescans
      s_mask[wp] = 0;
      myMax = 0ull;
      for (int p = p0; p < p1; ++p) {
        float v = s_conf[p];
        float s = (s_mask[p] && v > CONF_THR) ? v : NEGV;
        unsigned long long kk = ((unsigned long long)flip_f32(s) << 15) |
                                (unsigned long long)(P_PRIORS - p);
        if (kk > myMax) myMax = kk;
      }
    }
  }
  __syncthreads();

  // ---- decode the (<=100) surviving candidates: ARM refine -> ODM decode ----
  if (t < TOPK) {
    s_alive[t] = 1;
    float sc = s_selScore[t];
    float x1 = 0.f, y1 = 0.f, x2 = 0.f, y2 = 0.f;
    if (sc > CONF_THR) {
      int p = s_selIdx[t];
      const float* pr = prior + (size_t)p * 4;
      const float* al = bi_loc + ((size_t)b * P_PRIORS + p) * 4;
      const float* ml = multi_loc + ((size_t)b * P_PRIORS + p) * 4;
      float px = pr[0], py = pr[1], pw = pr[2], ph = pr[3];
      float cx = px + al[0] * 0.1f * pw;
      float cy = py + al[1] * 0.1f * ph;
      float rw = pw * expf(al[2] * 0.2f);
      float rh = ph * expf(al[3] * 0.2f);
      float ox = cx + ml[0] * 0.1f * rw;
      float oy = cy + ml[1] * 0.1f * rh;
      float ow = rw * expf(ml[2] * 0.2f);
      float oh = rh * expf(ml[3] * 0.2f);
      x1 = ox - 0.5f * ow; y1 = oy - 0.5f * oh;
      x2 = ox + 0.5f * ow; y2 = oy + 0.5f * oh;
    }
    s_box[t][0] = x1; s_box[t][1] = y1; s_box[t][2] = x2; s_box[t][3] = y2;
    s_area[t] = (x2 - x1) * (y2 - y1);
  }
  __syncthreads();

  // ---- greedy NMS (sequential 100-step scan, matches reference semantics) ----
  for (int step = 0; step < TOPK; ++step) {
    unsigned long long kk = 0ull;
    if (t < TOPK) {
      float s = s_alive[t] ? s_selScore[t] : NEGV;
      kk = ((unsigned long long)flip_f32(s) << 8) |
           (unsigned long long)(255 - t);        // tie -> lower index (argmax)
    }
    s_part[t] = kk;
    __syncthreads();
    for (int off = 128; off > 0; off >>= 1) {
      if (t < off) { if (s_part[t + off] > s_part[t]) s_part[t] = s_part[t + off]; }
      __syncthreads();
    }
    unsigned long long w = s_part[0];
    int sel = 255 - (int)(w & 0xFFull);
    float ss = unflip_f32((unsigned)(w >> 8));
    if (t == 0) {
      float* row = plane + (size_t)step * 5;
      if (ss > CONF_THR) {
        row[0] = ss;
        row[1] = s_box[sel][0]; row[2] = s_box[sel][1];
        row[3] = s_box[sel][2]; row[4] = s_box[sel][3];
      } else {
        row[0] = 0.f; row[1] = 0.f; row[2] = 0.f; row[3] = 0.f; row[4] = 0.f;
      }
    }
    __syncthreads();                    // w consumed before s_part reuse
    if (t < TOPK) {
      float tlx = fmaxf(s_box[sel][0], s_box[t][0]);
      float tly = fmaxf(s_box[sel][1], s_box[t][1]);
      float brx = fminf(s_box[sel][2], s_box[t][2]);
      float bry = fminf(s_box[sel][3], s_box[t][3]);
      float iw = fmaxf(brx - tlx, 0.0f);
      float ih = fmaxf(bry - tly, 0.0f);
      float inter = iw * ih;
      float uni = s_area[sel] + s_area[t] - inter;
      float iou = inter / fmaxf(uni, 1e-12f);
      if (!(iou <= NMS_THR) || t == sel) s_alive[t] = 0;
    }
    __syncthreads();
  }
}

// ---------- host launcher ----------
extern "C" void kernel_launch(void* const* d_in, const int* in_sizes, int n_in,
                              void* d_out, int out_size, void* d_ws, size_t ws_size,
                              hipStream_t stream) {
  const float* bi_loc     = (const float*)d_in[0];
  const float* bi_conf    = (const float*)d_in[1];
  const float* multi_loc  = (const float*)d_in[2];
  const float* multi_conf = (const float*)d_in[3];
  const float* prior      = (const float*)d_in[4];
  float* out = (float*)d_out;
  unsigned char* mask = (unsigned char*)d_ws;   // B*P = 408000 bytes

  int n = BATCH * P_PRIORS;
  arm_mask_kernel<<<(n + 255) / 256, 256, 0, stream>>>(bi_conf, mask, n);

  dim3 grid(NUM_CLASSES, BATCH);                // 336 blocks, one per (b,c)
  detect_topk_nms_kernel<<<grid, 256, 0, stream>>>(bi_loc, multi_loc, multi_conf,
                                                   prior, mask, out);
}